// FullEncoder_4226247819610
// MI455X (gfx1250) — compile-verified
//
#include <hip/hip_runtime.h>
#include <hip/hip_bf16.h>
#include <math.h>
#include <float.h>

typedef __attribute__((ext_vector_type(2))) float v2f;
typedef __attribute__((ext_vector_type(8))) float v8f;

#define KNN_K 16
#define WAVES_PER_BLOCK 8

// -------------------------------------------------------------------------
// Kernel 1: per-node feature build + tiny MLP encode (memory-bound, trivial)
// -------------------------------------------------------------------------
__global__ void feat_encode_kernel(const float* __restrict__ coords,
                                   const float* __restrict__ demands,
                                   const float* __restrict__ capacity,
                                   const float* __restrict__ cur,
                                   const float* __restrict__ Wa,
                                   const float* __restrict__ ba,
                                   const float* __restrict__ W1,
                                   const float* __restrict__ b1,
                                   const float* __restrict__ W2,
                                   const float* __restrict__ b2,
                                   float* __restrict__ psi_out,
                                   float* __restrict__ feat_out,
                                   int Bn, int Nn) {
    int gid = blockIdx.x * blockDim.x + threadIdx.x;
    if (gid >= Bn * Nn) return;
    int b = gid / Nn;

    float x  = coords[(size_t)gid * 2 + 0];
    float y  = coords[(size_t)gid * 2 + 1];
    float dx = coords[(size_t)b * Nn * 2 + 0];   // depot = node 0
    float dy = coords[(size_t)b * Nn * 2 + 1];
    float rx = x - dx, ry = y - dy;
    float dist_depot = sqrtf(rx * rx + ry * ry);
    float angle = atan2f(ry, rx) * 0.3183098861837907f;  // /pi
    float dem = demands[gid] / capacity[b];
    float cx = cur[b * 2 + 0], cy = cur[b * 2 + 1];
    float ux = x - cx, uy = y - cy;
    float dist_cur = sqrtf(ux * ux + uy * uy);

    float f[6] = {x, y, dem, dist_depot, angle, dist_cur};
    float2* fo = (float2*)(feat_out + (size_t)gid * 6);  // 8B aligned
    fo[0] = make_float2(f[0], f[1]);
    fo[1] = make_float2(f[2], f[3]);
    fo[2] = make_float2(f[4], f[5]);

    // psi = f @ Wa.T + ba ; normalize
    float p0 = ba[0], p1 = ba[1];
    #pragma unroll
    for (int k = 0; k < 6; ++k) { p0 += Wa[k] * f[k]; p1 += Wa[6 + k] * f[k]; }
    float inv = 1.0f / (sqrtf(p0 * p0 + p1 * p1) + 1e-8f);
    p0 *= inv; p1 *= inv;

    // theta = tanh(f @ W1.T + b1) @ W2.T + b2
    float theta = b2[0];
    #pragma unroll
    for (int h = 0; h < 16; ++h) {
        float a = b1[h];
        #pragma unroll
        for (int k = 0; k < 6; ++k) a += W1[h * 6 + k] * f[k];
        theta += W2[h] * tanhf(a);
    }
    float s, c;
    __sincosf(theta, &s, &c);
    *(float2*)(psi_out + (size_t)gid * 2) = make_float2(c * p0 - s * p1,
                                                        s * p0 + c * p1);
}

// -------------------------------------------------------------------------
// Register top-k kept sorted DESCENDING (best[0] = current worst).
// Insert = overwrite slot 0 + one branchless bubble pass (cndmask chain).
// -------------------------------------------------------------------------
__device__ __forceinline__ void topk_push(float (&best)[KNN_K], int (&bidx)[KNN_K],
                                          float d2, int j) {
    if (d2 < best[0]) {
        best[0] = d2; bidx[0] = j;
        #pragma unroll
        for (int t = 0; t + 1 < KNN_K; ++t) {
            float a = best[t], b = best[t + 1];
            bool sw = a < b;                  // out of descending order -> swap
            best[t]     = sw ? b : a;
            best[t + 1] = sw ? a : b;
            int ia = bidx[t], ib = bidx[t + 1];
            bidx[t]     = sw ? ib : ia;
            bidx[t + 1] = sw ? ia : ib;
        }
    }
}

// -------------------------------------------------------------------------
// Kernel 2: fused KNN. One wave per 16-row tile.
// Cross terms via V_WMMA_F32_16X16X4_F32 (K padded 2->4 with zeros, -2 folded
// into A). D tile transposed through LDS; 2 lanes/row keep register top-16.
// Coords load software-pipelined one tile ahead; diagonal mask hoisted to the
// single (wave-uniform) jt == rt tile.
// -------------------------------------------------------------------------
__global__ void knn_wmma_kernel(const float* __restrict__ coords,
                                int* __restrict__ knn_out,
                                int Bn, int Nn) {
    __shared__ float lds_dot[WAVES_PER_BLOCK][16][20];  // stride 20 -> 16B-aligned rows
    __shared__ float lds_sqc[WAVES_PER_BLOCK][16];

    const int w    = threadIdx.x >> 5;
    const int lane = threadIdx.x & 31;
    const int r16  = lane & 15;
    const int half = lane >> 4;

    const int tilesPerBatch = Nn >> 4;
    const int tile = blockIdx.x * WAVES_PER_BLOCK + w;
    if (tile >= Bn * tilesPerBatch) return;          // wave-uniform branch
    const int b  = tile / tilesPerBatch;
    const int rt = tile - b * tilesPerBatch;
    const int r0 = rt << 4;

    const float* cb = coords + (size_t)b * Nn * 2;

    // Query row owned by this lane pair
    const int grow = r0 + r16;
    const float qx = cb[(size_t)grow * 2 + 0];
    const float qy = cb[(size_t)grow * 2 + 1];
    const float sq_q = qx * qx + qy * qy;

    // A (16x4 f32): lanes 0-15 hold K=0,1 = (-2x,-2y); lanes 16-31 hold K=2,3 = 0
    v2f amat;
    amat.x = half ? 0.0f : (-2.0f * qx);
    amat.y = half ? 0.0f : (-2.0f * qy);

    float best[KNN_K]; int bidx[KNN_K];
    #pragma unroll
    for (int t = 0; t < KNN_K; ++t) { best[t] = FLT_MAX; bidx[t] = -1; }

    // Software pipeline: candidate coords for tile jt loaded one iter ahead
    float2 cc = *(const float2*)(cb + (size_t)r16 * 2);   // tile 0

    for (int jt = 0; jt < tilesPerBatch; ++jt) {
        const int j0 = jt << 4;
        // Issue next tile's load now; waited on only at next iteration's use
        const int jn = (jt + 1 < tilesPerBatch) ? (jt + 1) : 0;
        float2 cc_next = *(const float2*)(cb + (size_t)((jn << 4) + r16) * 2);

        // B (4x16 f32): lanes 0-15 hold K=0,1 = (x,y); lanes 16-31 hold K=2,3 = 0
        v2f bmat;
        bmat.x = half ? 0.0f : cc.x;
        bmat.y = half ? 0.0f : cc.y;
        lds_sqc[w][r16] = cc.x * cc.x + cc.y * cc.y;   // both halves: same value

        v8f acc = {};
        acc = __builtin_amdgcn_wmma_f32_16x16x4_f32(false, amat, false, bmat,
                                                    (short)0, acc, false, false);

        // D layout: lane L, vgpr v -> (M = v + 8*(L>=16), N = L%16)
        #pragma unroll
        for (int v = 0; v < 8; ++v)
            lds_dot[w][v + 8 * half][r16] = acc[v];

        asm volatile("s_wait_dscnt 0" ::: "memory");

        // Selection: lane (r16,half) scans row r16, columns half*8 .. half*8+7
        const float* drow = &lds_dot[w][r16][0];
        const float4 dA = *(const float4*)(drow + (half << 3));
        const float4 dB = *(const float4*)(drow + (half << 3) + 4);
        const float4 sA = *(const float4*)(&lds_sqc[w][half << 3]);
        const float4 sB = *(const float4*)(&lds_sqc[w][(half << 3) + 4]);
        const float dd[8] = {dA.x, dA.y, dA.z, dA.w, dB.x, dB.y, dB.z, dB.w};
        const float ss[8] = {sA.x, sA.y, sA.z, sA.w, sB.x, sB.y, sB.z, sB.w};

        float d2v[8];
        #pragma unroll
        for (int t = 0; t < 8; ++t)
            d2v[t] = sq_q + ss[t] + dd[t];          // sq_i + sq_j - 2*dot

        // Only the jt == rt tile touches the diagonal (wave-uniform test)
        if (jt == rt) {
            #pragma unroll
            for (int t = 0; t < 8; ++t)
                if (((half << 3) + t) == r16) d2v[t] = FLT_MAX;
        }

        // Tile-level early-out: one compare guards all 8 insert attempts
        float m = d2v[0];
        #pragma unroll
        for (int t = 1; t < 8; ++t) m = fminf(m, d2v[t]);
        if (m < best[0]) {
            const int jbase = j0 + (half << 3);
            #pragma unroll
            for (int t = 0; t < 8; ++t) topk_push(best, bidx, d2v[t], jbase + t);
        }
        cc = cc_next;
        asm volatile("" ::: "memory");
    }

    // Merge the half==1 partial lists into half==0's list
    #pragma unroll
    for (int t = 0; t < KNN_K; ++t) {
        float dv = __shfl_xor(best[t], 16, 32);
        int   iv = __shfl_xor(bidx[t], 16, 32);
        if (half == 0) topk_push(best, bidx, dv, iv);
    }

    if (half == 0) {
        // List is sorted descending by d2 -> emit ascending (top_k(-d2) order)
        int* op = knn_out + (size_t)(b * Nn + grow) * KNN_K;
        #pragma unroll
        for (int t = 0; t < KNN_K; t += 4) {
            int4 vv = make_int4(bidx[KNN_K - 1 - t], bidx[KNN_K - 2 - t],
                                bidx[KNN_K - 3 - t], bidx[KNN_K - 4 - t]);
            *(int4*)(op + t) = vv;
        }
    }
}

// -------------------------------------------------------------------------
extern "C" void kernel_launch(void* const* d_in, const int* in_sizes, int n_in,
                              void* d_out, int out_size, void* d_ws, size_t ws_size,
                              hipStream_t stream) {
    const float* coords   = (const float*)d_in[0];
    const float* demands  = (const float*)d_in[1];
    const float* capacity = (const float*)d_in[2];
    const float* cur      = (const float*)d_in[3];
    const float* Wa       = (const float*)d_in[4];
    const float* ba       = (const float*)d_in[5];
    const float* W1       = (const float*)d_in[6];
    const float* b1       = (const float*)d_in[7];
    const float* W2       = (const float*)d_in[8];
    const float* b2       = (const float*)d_in[9];
    (void)d_ws; (void)ws_size; (void)n_in;

    const int Bn    = in_sizes[2];             // capacity has B elements
    const int total = in_sizes[0] / 2;         // B*N nodes
    const int Nn    = total / Bn;

    float* psi_out  = (float*)d_out;                        // (B,N,2)
    float* feat_out = psi_out + (size_t)total * 2;          // (B,N,6)
    int*   knn_out  = (int*)(feat_out + (size_t)total * 6); // (B,N,16) int32 bits
    (void)out_size;

    {
        int threads = 256;
        int blocks  = (total + threads - 1) / threads;
        feat_encode_kernel<<<blocks, threads, 0, stream>>>(
            coords, demands, capacity, cur, Wa, ba, W1, b1, W2, b2,
            psi_out, feat_out, Bn, Nn);
    }
    {
        int rowTiles = total / 16;                          // 2048
        int blocks   = (rowTiles + WAVES_PER_BLOCK - 1) / WAVES_PER_BLOCK;
        knn_wmma_kernel<<<blocks, 32 * WAVES_PER_BLOCK, 0, stream>>>(
            coords, knn_out, Bn, Nn);
    }
}